// Loss_Reg_47184510714648
// MI455X (gfx1250) — compile-verified
//
#include <hip/hip_runtime.h>
#include <hip/hip_bf16.h>

// Problem constants (B=32, N=1024 from the reference)
constexpr int NB   = 32;     // batches
constexpr int NN   = 1024;   // matrix dim
constexpr int NSEG = NN - 2; // segments actually used: k = 0..1021 (cnt >= 2)
constexpr int CH   = 128;    // i-chunk length per thread in kernel A
constexpr int NC   = 8;      // number of i-chunks (8*128 = 1024 >= 1023)

typedef __attribute__((ext_vector_type(2))) float v2f;
typedef __attribute__((ext_vector_type(8))) float v8f;

// ---------------------------------------------------------------------------
// Kernel A: per-(batch, segment, i-chunk) partial sums along diagonals.
// Wave lanes own 32 consecutive segments k -> loads of 32 consecutive floats
// (fully coalesced); per-lane register accumulation, no atomics.
// ---------------------------------------------------------------------------
__global__ __launch_bounds__(256)
void seg_partials(const float* __restrict__ S,
                  float* __restrict__ wsum,
                  float* __restrict__ wsq) {
    const int k = blockIdx.x * blockDim.x + threadIdx.x;   // segment index
    if (k >= NSEG) return;
    const int c = blockIdx.y;                              // i-chunk
    const int b = blockIdx.z;                              // batch

    const int cnt = NN - 1 - k;                            // diagonal length
    const int i0  = c * CH;
    const int i1  = (cnt < i0 + CH) ? cnt : (i0 + CH);

    float s = 0.0f, q = 0.0f;
    if (i1 > i0) {
        // element (i, i+k+1): flat index b*N*N + i*(N+1) + (k+1)
        const float* p = S + (size_t)b * NN * NN
                           + (size_t)i0 * (NN + 1) + (size_t)(k + 1);
        // Speculative prefetch of the next chunk's first line
        // (lowers to global_prefetch_b8 on gfx1250; dropped if invalid).
        __builtin_prefetch(p + (size_t)CH * (NN + 1), 0, 0);

        const int n = i1 - i0;
        if (n == CH) {
            #pragma unroll
            for (int t = 0; t < CH; t += 16) {
                float v[16];
                #pragma unroll
                for (int u = 0; u < 16; ++u)
                    v[u] = p[(size_t)(t + u) * (NN + 1)];   // 16 loads in flight
                #pragma unroll
                for (int u = 0; u < 16; ++u) {
                    s += v[u];
                    q = fmaf(v[u], v[u], q);
                }
            }
        } else {
            for (int t = 0; t < n; ++t) {
                float v = p[(size_t)t * (NN + 1)];
                s += v;
                q = fmaf(v, v, q);
            }
        }
    }
    const size_t idx = ((size_t)(b * NSEG + k)) * NC + (size_t)c;
    wsum[idx] = s;
    wsq[idx]  = q;
}

// ---------------------------------------------------------------------------
// Exact 32-lane f32 wave sum using V_WMMA_F32_16X16X4_F32.
// A (16x4 f32) layout: VGPR0 holds {lanes 0-15 -> (M=lane, K=0),
// lanes 16-31 -> (M=lane-16, K=2)}; we place v there and zero VGPR1, so with
// B = all-ones: D[m,n] = v[m] + v[m+16] for every n.  D's 8 VGPRs per lane:
// lanes<16 carry rows 0..7, lanes>=16 rows 8..15.  Summing the 8 VGPRs and
// folding halves with shfl_xor(16) yields the full 32-lane sum in f32.
// Requires EXEC all-ones: call only from fully-converged, wave-uniform code.
// ---------------------------------------------------------------------------
__device__ __forceinline__ float wave_sum32_wmma(float x) {
    v2f a; a.x = x;    a.y = 0.0f;
    v2f o; o.x = 1.0f; o.y = 1.0f;
    v8f acc = {};
    acc = __builtin_amdgcn_wmma_f32_16x16x4_f32(
        /*neg_a=*/false, a, /*neg_b=*/false, o,
        /*c_mod=*/(short)0, acc, /*reuse_a=*/false, /*reuse_b=*/false);
    float s = acc[0] + acc[1] + acc[2] + acc[3]
            + acc[4] + acc[5] + acc[6] + acc[7];
    s += __shfl_xor(s, 16, 32);
    return s;
}

// ---------------------------------------------------------------------------
// Kernel B: combine chunk partials -> std_scaled per (b,k), deterministic
// block-wide reduction to the final scalar.
// ---------------------------------------------------------------------------
__global__ __launch_bounds__(1024)
void seg_finalize(const float* __restrict__ wsum,
                  const float* __restrict__ wsq,
                  float* __restrict__ out) {
    __shared__ float red[32];
    const int tid = threadIdx.x;

    float accum = 0.0f;
    const int P = NB * NSEG;
    for (int p = tid; p < P; p += 1024) {
        const int b = p / NSEG;
        const int k = p - b * NSEG;
        const size_t base = ((size_t)(b * NSEG + k)) * NC;
        float s = 0.0f, q = 0.0f;
        #pragma unroll
        for (int c = 0; c < NC; ++c) {
            s += wsum[base + c];
            q += wsq[base + c];
        }
        const float cnt  = (float)(NN - 1 - k);
        const float mean = s / cnt;
        const float var  = (q - cnt * mean * mean) / (cnt - 1.0f);
        accum += sqrtf(fmaxf(var, 0.0f)) * cnt * 0.05f;   // *cnt/20
    }

    // wave-level sum (all 1024 threads converged here -> EXEC all ones)
    const float wsumv = wave_sum32_wmma(accum);
    const int lane = tid & 31;
    const int wid  = tid >> 5;
    if (lane == 0) red[wid] = wsumv;
    __syncthreads();

    if (wid == 0) {                       // wave-uniform branch: EXEC all ones
        const float v   = red[lane];      // 32 wave partials
        const float tot = wave_sum32_wmma(v);
        if (lane == 0)
            out[0] = tot / (float)(NB * NSEG);   // /(32*1022): both means
    }
}

// ---------------------------------------------------------------------------
extern "C" void kernel_launch(void* const* d_in, const int* in_sizes, int n_in,
                              void* d_out, int out_size, void* d_ws, size_t ws_size,
                              hipStream_t stream) {
    (void)in_sizes; (void)n_in; (void)out_size; (void)ws_size;
    const float* S = (const float*)d_in[0];

    float* wsum = (float*)d_ws;                         // NB*NSEG*NC floats
    float* wsq  = wsum + (size_t)NB * NSEG * NC;        // NB*NSEG*NC floats
    // total workspace: 2 * 32 * 1022 * 8 * 4 B ≈ 2.09 MB

    dim3 gridA((NN + 255) / 256, NC, NB);   // 4 x 8 x 32 = 1024 blocks
    dim3 blockA(256);
    seg_partials<<<gridA, blockA, 0, stream>>>(S, wsum, wsq);

    seg_finalize<<<1, 1024, 0, stream>>>(wsum, wsq, (float*)d_out);
}